// PeriodicBoundary_21320217657568
// MI455X (gfx1250) — compile-verified
//
#include <hip/hip_runtime.h>

// CDNA5 / gfx1250: wave32, WMMA f32 16x16x4 for the K=3 (padded to 4) dot-product GEMM
typedef __attribute__((ext_vector_type(2))) float v2f;
typedef __attribute__((ext_vector_type(8))) float v8f;

#define NPTS   1024
#define NG     27
#define GN     (NG * NPTS)   // 27648 image points
#define NTILES (GN / 16)     // 1728 column tiles
#define MAXN   128
#define CUT2   25.0f

// One block = one wave32 = one 16-row M-tile of centers.
// Scans all 1728 16-column N-tiles in ascending j order so the per-row
// compaction matches jnp.argwhere's ordering.
__global__ __launch_bounds__(32)
void pb_neigh_kernel(const float* __restrict__ pos,     // (1024,3)
                     const float* __restrict__ grid,    // (27,3) float shifts
                     const int*   __restrict__ cl,      // (27,3) int shifts
                     int* __restrict__ dneigh,          // (1024,128)
                     int* __restrict__ dcell,           // (1024,128,3)
                     int* __restrict__ wsmax)           // (64,) per-block max count
{
    __shared__ float    sh_pos[NPTS * 3];
    __shared__ float    sh_grid[NG * 3];
    __shared__ int      sh_cell[NG * 3];
    __shared__ int      sh_cnt[16];

    const int lane  = threadIdx.x;      // 0..31
    const int mtile = blockIdx.x;       // 0..63
    const int hi    = lane >> 4;        // half-wave select
    const int lo    = lane & 15;

    for (int k = lane; k < NPTS * 3; k += 32) sh_pos[k] = pos[k];
    for (int k = lane; k < NG * 3;  k += 32) { sh_grid[k] = grid[k]; sh_cell[k] = cl[k]; }
    __syncthreads();

    // A fragment (16x4 f32): lanes 0-15 -> K=0,1 (x,y); lanes 16-31 -> K=2,3 (z,0)
    v2f afrag;
    {
        const int ig = mtile * 16 + lo;
        const float x = sh_pos[ig * 3 + 0];
        const float y = sh_pos[ig * 3 + 1];
        const float z = sh_pos[ig * 3 + 2];
        afrag[0] = hi ? z : x;
        afrag[1] = hi ? 0.0f : y;
    }

    // |p|^2 for the 8 C/D rows this lane holds (D layout: VGPR v -> row v + 8*hi)
    float rp2row[8];
#pragma unroll
    for (int v = 0; v < 8; ++v) {
        const int ig = mtile * 16 + v + 8 * hi;
        const float x = sh_pos[ig * 3 + 0];
        const float y = sh_pos[ig * 3 + 1];
        const float z = sh_pos[ig * 3 + 2];
        rp2row[v] = x * x + y * y + z * z;
    }

    int cnt = 0;                               // unclipped hit count for row `lo`
    const int iglob = mtile * 16 + lo;

    for (int t = 0; t < NTILES; ++t) {
        // Image point for this lane's column: q = pos[j % 1024] + grid[j / 1024]
        const int j  = t * 16 + lo;
        const int jm = j & (NPTS - 1);
        const int jg = j >> 10;
        const float qx = sh_pos[jm * 3 + 0] + sh_grid[jg * 3 + 0];
        const float qy = sh_pos[jm * 3 + 1] + sh_grid[jg * 3 + 1];
        const float qz = sh_pos[jm * 3 + 2] + sh_grid[jg * 3 + 2];
        const float rq2 = qx * qx + qy * qy + qz * qz;

        // B fragment (4x16 f32): lanes 0-15 -> K=0,1 (x,y); lanes 16-31 -> K=2,3 (z,0)
        v2f bfrag;
        bfrag[0] = hi ? qz : qx;
        bfrag[1] = hi ? 0.0f : qy;

        v8f c = {};
        c = __builtin_amdgcn_wmma_f32_16x16x4_f32(false, afrag, false, bfrag,
                                                  (short)0, c, false, false);

        // Per-C-VGPR ballot -> wave-uniform 32-bit masks: bits [15:0] = row v,
        // bits [31:16] = row v+8 (D layout: N = lane%16).
        unsigned bal[8];
#pragma unroll
        for (int v = 0; v < 8; ++v) {
            const float d2 = rp2row[v] + rq2 - 2.0f * c[v];
            bal[v] = __builtin_amdgcn_ballot_w32(d2 < CUT2);
        }

        // Lane r (<16) owns row r: pick its mask with a cndmask select chain
        // (bal[] is uniform/SGPR -> no exec-mask gymnastics, no LDS round-trip).
        unsigned m = 0;
#pragma unroll
        for (int v = 0; v < 8; ++v)
            m |= ((lo & 7) == v) ? bal[v] : 0u;
        m = (lo < 8) ? (m & 0xFFFFu) : (m >> 16);
        if (t == mtile) m &= ~(1u << lo);  // faithful self-mask (j == i); numerically a no-op

        // Ordered compaction (ascending j within the tile)
        if (lane < 16) {
            while (m) {
                const int n = __builtin_ctz(m);
                m &= m - 1;
                const int jc = t * 16 + n;     // flat image index, ascending
                if (cnt < MAXN) {
                    const int o = iglob * MAXN + cnt;
                    dneigh[o] = jc & (NPTS - 1);
                    const int g = (jc >> 10) * 3;
                    dcell[o * 3 + 0] = sh_cell[g + 0];
                    dcell[o * 3 + 1] = sh_cell[g + 1];
                    dcell[o * 3 + 2] = sh_cell[g + 2];
                }
                cnt++;
            }
        }
    }

    // Padding: neigh = -1; cell pad = cells[-1] = cell_list[26] (negative-index wrap)
    if (lane < 16) {
        const int f0 = sh_cell[(NG - 1) * 3 + 0];
        const int f1 = sh_cell[(NG - 1) * 3 + 1];
        const int f2 = sh_cell[(NG - 1) * 3 + 2];
        for (int k = (cnt < MAXN ? cnt : MAXN); k < MAXN; ++k) {
            const int o = iglob * MAXN + k;
            dneigh[o] = -1;
            dcell[o * 3 + 0] = f0;
            dcell[o * 3 + 1] = f1;
            dcell[o * 3 + 2] = f2;
        }
        sh_cnt[lo] = cnt;
    }
    __syncthreads();
    if (lane == 0) {
        int mx = 0;
        for (int k = 0; k < 16; ++k) mx = sh_cnt[k] > mx ? sh_cnt[k] : mx;
        wsmax[mtile] = mx;
    }
}

__global__ void pb_max_kernel(const int* __restrict__ wsmax, int* __restrict__ amax)
{
    if (threadIdx.x == 0) {
        int mx = 0;
        for (int k = 0; k < 64; ++k) mx = wsmax[k] > mx ? wsmax[k] : mx;
        amax[0] = mx;
    }
}

extern "C" void kernel_launch(void* const* d_in, const int* in_sizes, int n_in,
                              void* d_out, int out_size, void* d_ws, size_t ws_size,
                              hipStream_t stream)
{
    (void)in_sizes; (void)n_in; (void)out_size; (void)ws_size;

    const float* pos  = (const float*)d_in[0];   // (1024,3) f32
    const float* grid = (const float*)d_in[1];   // (27,3)   f32
    const int*   cl   = (const int*)  d_in[2];   // (27,3)   i32
    // d_in[3] = max_neighbours (== 128, baked into MAXN)

    int* out    = (int*)d_out;
    int* dneigh = out;                                    // 1024*128
    int* dcell  = out + NPTS * MAXN;                      // 1024*128*3
    int* amax   = out + NPTS * MAXN + NPTS * MAXN * 3;    // 1
    int* wsmax  = (int*)d_ws;                             // 64 ints

    pb_neigh_kernel<<<64, 32, 0, stream>>>(pos, grid, cl, dneigh, dcell, wsmax);
    pb_max_kernel<<<1, 32, 0, stream>>>(wsmax, amax);
}